// MyGraphModel_9474697855312
// MI455X (gfx1250) — compile-verified
//
#include <hip/hip_runtime.h>

typedef __attribute__((ext_vector_type(2))) float v2f;
typedef __attribute__((ext_vector_type(8))) float v8f;

#define DF 64
#define LDS_STRIDE 68   // 64 + 4 pad: B-fragment reads hit all 64 banks conflict-free

// ---------------------------------------------------------------------------
// Scatter: agg[dst] += x[src] (64 floats), cnt[dst] += 1 (first pass only).
// One 32-lane slice of threads per edge; lane handles 2 contiguous floats so
// the x[src] row read is a coalesced 256B transaction per edge.
// ---------------------------------------------------------------------------
__global__ void scatter_add(const float* __restrict__ xin,
                            const int* __restrict__ src,
                            const int* __restrict__ dst,
                            float* __restrict__ agg,
                            float* __restrict__ cnt,
                            int n_edges, int do_cnt) {
  long long t = (long long)blockIdx.x * blockDim.x + threadIdx.x;
  int e    = (int)(t >> 5);
  int lane = (int)(t & 31);
  if (e >= n_edges) return;
  int s = src[e];
  int d = dst[e];
  float2 v = *(const float2*)(xin + (long long)s * DF + lane * 2);
  float* ar = agg + (long long)d * DF + lane * 2;
  atomicAdd(ar,     v.x);
  atomicAdd(ar + 1, v.y);
  if (do_cnt && lane == 0) atomicAdd(cnt + d, 1.0f);
}

// ---------------------------------------------------------------------------
// Fused SAGE linear: out = [relu]( (agg/max(cnt,1)) @ Wl^T + bl + xin @ Wr^T )
// One wave per 16-node tile. WMMA f32 16x16x4:
//   A lane l: row = l&15, K = 2*(l>>4) + vgpr          (16x4 A tile)
//   B lane l: col n = l&15, K = 2*(l>>4) + vgpr  -> reads W row-major [n][k]
//   C/D vgpr g: row = g + 8*(l>>4), col = l&15
// ---------------------------------------------------------------------------
__global__ void __launch_bounds__(128)
sage_linear(const float* __restrict__ agg,
            const float* __restrict__ cnt,
            const float* __restrict__ xin,
            const float* __restrict__ Wl,   // [64 x 64] row-major (out x in)
            const float* __restrict__ bl,   // [64]
            const float* __restrict__ Wr,   // [64 x 64]
            float* __restrict__ out,
            int n_nodes, int do_relu) {
  __shared__ float sWl[DF * LDS_STRIDE];
  __shared__ float sWr[DF * LDS_STRIDE];

  // Cooperative weight stage: 2048 float2 per matrix across 128 threads.
  for (int i = threadIdx.x; i < DF * DF / 2; i += 128) {
    int r = i >> 5;
    int c = (i & 31) * 2;
    *(float2*)&sWl[r * LDS_STRIDE + c] = *(const float2*)&Wl[r * DF + c];
    *(float2*)&sWr[r * LDS_STRIDE + c] = *(const float2*)&Wr[r * DF + c];
  }
  __syncthreads();

  int wave = threadIdx.x >> 5;
  int lane = threadIdx.x & 31;
  int tile = blockIdx.x * 4 + wave;
  int nb   = tile * 16;
  if (nb >= n_nodes) return;     // wave-uniform: EXEC stays all-ones for WMMA

  int m  = lane & 15;            // A row / B col / D col
  int kh = lane >> 4;            // K-half select

  int row = nb + m;
  float invc = 1.0f / fmaxf(cnt[row], 1.0f);
  const float* arow = agg + (long long)row * DF + 2 * kh;
  const float* xrow = xin + (long long)row * DF + 2 * kh;

  v8f acc[4];
#pragma unroll
  for (int nt = 0; nt < 4; ++nt) {
    float b = bl[nt * 16 + m];   // bias depends only on output column
    v8f c = {b, b, b, b, b, b, b, b};
    acc[nt] = c;
  }

  for (int k = 0; k < DF; k += 4) {
    v2f am = *(const v2f*)(arow + k);
    am *= invc;                              // mean = agg / max(cnt,1)
    v2f ax = *(const v2f*)(xrow + k);
#pragma unroll
    for (int nt = 0; nt < 4; ++nt) {
      v2f bw = *(const v2f*)&sWl[(nt * 16 + m) * LDS_STRIDE + k + 2 * kh];
      acc[nt] = __builtin_amdgcn_wmma_f32_16x16x4_f32(
          false, am, false, bw, (short)0, acc[nt], false, false);
      v2f br = *(const v2f*)&sWr[(nt * 16 + m) * LDS_STRIDE + k + 2 * kh];
      acc[nt] = __builtin_amdgcn_wmma_f32_16x16x4_f32(
          false, ax, false, br, (short)0, acc[nt], false, false);
    }
  }

#pragma unroll
  for (int nt = 0; nt < 4; ++nt) {
#pragma unroll
    for (int g = 0; g < 8; ++g) {
      int orow = nb + g + 8 * kh;
      float v = acc[nt][g];
      if (do_relu) v = fmaxf(v, 0.0f);
      out[(long long)orow * DF + nt * 16 + m] = v;
    }
  }
}

// ---------------------------------------------------------------------------
extern "C" void kernel_launch(void* const* d_in, const int* in_sizes, int n_in,
                              void* d_out, int out_size, void* d_ws, size_t ws_size,
                              hipStream_t stream) {
  const float* x   = (const float*)d_in[0];
  const int*   ei  = (const int*)d_in[1];
  const float* W1l = (const float*)d_in[2];
  const float* b1l = (const float*)d_in[3];
  const float* W1r = (const float*)d_in[4];
  const float* W2l = (const float*)d_in[5];
  const float* b2l = (const float*)d_in[6];
  const float* W2r = (const float*)d_in[7];
  float* out = (float*)d_out;

  int n_nodes = in_sizes[0] / DF;      // 100000
  int n_edges = in_sizes[1] / 2;       // 1600000
  const int* src = ei;
  const int* dst = ei + n_edges;

  size_t aggBytes = (size_t)n_nodes * DF * sizeof(float);
  char* ws = (char*)d_ws;
  float* agg = (float*)ws;                        // N*64 f32
  float* h   = (float*)(ws + aggBytes);           // N*64 f32 (layer-1 output)
  float* cnt = (float*)(ws + 2 * aggBytes);       // N f32

  long long etot = (long long)n_edges * 32;
  int eblocks = (int)((etot + 255) / 256);
  int tiles   = (n_nodes + 15) / 16;
  int gblocks = (tiles + 3) / 4;

  // ---- layer 1 ----
  hipMemsetAsync(agg, 0, aggBytes, stream);
  hipMemsetAsync(cnt, 0, (size_t)n_nodes * sizeof(float), stream);
  scatter_add<<<eblocks, 256, 0, stream>>>(x, src, dst, agg, cnt, n_edges, 1);
  sage_linear<<<gblocks, 128, 0, stream>>>(agg, cnt, x, W1l, b1l, W1r, h,
                                           n_nodes, 1);

  // ---- layer 2 (cnt is reusable: depends only on edge structure) ----
  hipMemsetAsync(agg, 0, aggBytes, stream);
  scatter_add<<<eblocks, 256, 0, stream>>>(h, src, dst, agg, cnt, n_edges, 0);
  sage_linear<<<gblocks, 128, 0, stream>>>(agg, cnt, h, W2l, b2l, W2r, out,
                                           n_nodes, 0);
}